// LandmarkGNN_52295521796621
// MI455X (gfx1250) — compile-verified
//
#include <hip/hip_runtime.h>

#define HID 64

typedef __attribute__((ext_vector_type(2))) float v2f;
typedef __attribute__((ext_vector_type(8))) float v8f;

// ---------------- degree / normalization ----------------

__global__ void k_deg_init(unsigned int* deg, int N) {
    int i = blockIdx.x * blockDim.x + threadIdx.x;
    if (i < N) deg[i] = 1u;  // self-loop contributes 1 to every node's degree
}

__global__ void k_deg_count(unsigned int* deg, const int* __restrict__ dst, int E) {
    int e = blockIdx.x * blockDim.x + threadIdx.x;
    if (e < E) atomicAdd(&deg[dst[e]], 1u);
}

__global__ void k_dinv(const unsigned int* __restrict__ deg, float* __restrict__ dinv, int N) {
    int i = blockIdx.x * blockDim.x + threadIdx.x;
    if (i < N) dinv[i] = rsqrtf((float)deg[i]);  // deg >= 1 always
}

// ---------------- layer 1: x@W1 (K=2) + self-loop init ----------------

__global__ void k_l1(const float* __restrict__ x, const float* __restrict__ W1,
                     const float* __restrict__ dinv,
                     float* __restrict__ h, float* __restrict__ agg, int N) {
    int t = blockIdx.x * blockDim.x + threadIdx.x;
    if (t >= N * HID) return;
    int i = t >> 6, j = t & 63;
    float v = x[2 * i] * W1[j] + x[2 * i + 1] * W1[HID + j];
    h[t] = v;
    float di = dinv[i];
    agg[t] = v * di * di;  // self-loop message
}

// ---------------- edge scatter: agg[dst] += h[src] * dinv[s]*dinv[d] ----------------
// blockDim = (64, 4): 64 lanes = feature dim (coalesced 256B gather per edge),
// 4 edges per block. f32 adds use hardware global_atomic_add_f32 (L2-resident).

__global__ __launch_bounds__(256) void k_scatter(const float* __restrict__ h,
                                                 float* __restrict__ agg,
                                                 const int* __restrict__ src,
                                                 const int* __restrict__ dst,
                                                 const float* __restrict__ dinv, int E) {
    int e = blockIdx.x * 4 + threadIdx.y;
    if (e >= E) return;
    int s = src[e], d = dst[e];
    float norm = dinv[s] * dinv[d];
    int j = threadIdx.x;
    unsafeAtomicAdd(&agg[(size_t)d * HID + j], h[(size_t)s * HID + j] * norm);
}

// ---------------- bias + relu ----------------

__global__ void k_bias_relu(const float* __restrict__ agg, const float* __restrict__ b,
                            float* __restrict__ out, int N) {
    int t = blockIdx.x * blockDim.x + threadIdx.x;
    if (t >= N * HID) return;
    float v = agg[t] + b[t & 63];
    out[t] = v > 0.0f ? v : 0.0f;
}

// ---------------- layer 2 GEMM: C[N x 64] = A[N x 64] @ B[64 x 64] via WMMA f32 ----------------
// One wave per 16x16 output tile; 4 waves/block cover n0 = 0,16,32,48 for one 16-row stripe.
// A 16x4 f32 fragment (2 VGPRs): lane holds row M = lane&15, K = k + 2*(lane>>4) + {0,1}.
// B 4x16 fragment mirrored: lane holds col N = lane&15, same K split.
// C/D 16x16 f32 (8 VGPRs): element (v + 8*(lane>>4), lane&15) in c[v].

__global__ __launch_bounds__(128) void k_gemm64(const float* __restrict__ A,
                                                const float* __restrict__ B,
                                                float* __restrict__ C, int N) {
    const int lane = threadIdx.x & 31;
    const int wid  = threadIdx.x >> 5;
    const int m0   = blockIdx.x * 16;
    const int n0   = wid * 16;
    const int mrow = lane & 15;
    const int koff = (lane >> 4) * 2;
    const int n    = lane & 15;

    int arow = m0 + mrow;
    if (arow >= N) arow = N - 1;  // clamp: keep EXEC all-1s for WMMA

    v8f c = {};
#pragma unroll
    for (int k = 0; k < HID; k += 4) {
        v2f a, b;
        a.x = A[(size_t)arow * HID + k + koff];
        a.y = A[(size_t)arow * HID + k + koff + 1];
        b.x = B[(k + koff) * HID + n0 + n];
        b.y = B[(k + koff + 1) * HID + n0 + n];
        c = __builtin_amdgcn_wmma_f32_16x16x4_f32(
                /*neg_a=*/false, a, /*neg_b=*/false, b,
                /*c_mod=*/(short)0, c, /*reuse_a=*/false, /*reuse_b=*/false);
    }

    const int rbase = m0 + (lane >> 4) * 8;
#pragma unroll
    for (int v = 0; v < 8; ++v) {
        int r = rbase + v;
        if (r < N) C[(size_t)r * HID + n0 + n] = c[v];
    }
}

// ---------------- layer 2 self-loop init ----------------

__global__ void k_selfloop(const float* __restrict__ h, const float* __restrict__ dinv,
                           float* __restrict__ agg, int N) {
    int t = blockIdx.x * blockDim.x + threadIdx.x;
    if (t >= N * HID) return;
    float di = dinv[t >> 6];
    agg[t] = h[t] * di * di;
}

// ---------------- pooling ----------------

__global__ void k_zero_pool(float* pooled, unsigned int* cnt, int G) {
    int t = blockIdx.x * blockDim.x + threadIdx.x;
    if (t < G * HID) pooled[t] = 0.0f;
    if (t < G) cnt[t] = 0u;
}

__global__ void k_relu_pool(const float* __restrict__ agg, const float* __restrict__ b,
                            const int* __restrict__ batch,
                            float* __restrict__ pooled, unsigned int* __restrict__ cnt, int N) {
    int t = blockIdx.x * blockDim.x + threadIdx.x;
    if (t >= N * HID) return;
    int i = t >> 6, j = t & 63;
    float v = agg[t] + b[j];
    v = v > 0.0f ? v : 0.0f;
    int g = batch[i];
    unsafeAtomicAdd(&pooled[(size_t)g * HID + j], v);
    if (j == 0) atomicAdd(&cnt[g], 1u);
}

__global__ void k_head(const float* __restrict__ pooled, const unsigned int* __restrict__ cnt,
                       const float* __restrict__ Wfc, const float* __restrict__ bfc,
                       float* __restrict__ out, int G) {
    int g = blockIdx.x * blockDim.x + threadIdx.x;
    if (g >= G) return;
    float acc = 0.0f;
#pragma unroll
    for (int j = 0; j < HID; ++j) acc += pooled[(size_t)g * HID + j] * Wfc[j];
    float c = (float)cnt[g];
    if (c < 1.0f) c = 1.0f;
    out[g] = acc / c + bfc[0];
}

// ---------------- launcher ----------------

extern "C" void kernel_launch(void* const* d_in, const int* in_sizes, int n_in,
                              void* d_out, int out_size, void* d_ws, size_t ws_size,
                              hipStream_t stream) {
    const float* x    = (const float*)d_in[0];
    const int*   ei   = (const int*)d_in[1];
    const int*   batch= (const int*)d_in[2];
    const float* W1   = (const float*)d_in[3];
    const float* b1   = (const float*)d_in[4];
    const float* W2   = (const float*)d_in[5];
    const float* b2   = (const float*)d_in[6];
    const float* Wfc  = (const float*)d_in[7];
    const float* bfc  = (const float*)d_in[8];
    float* out = (float*)d_out;

    const int N = in_sizes[0] / 2;   // x is [N, 2]
    const int E = in_sizes[1] / 2;   // edge_index is [2, E]
    const int G = out_size;          // [G, 1] output
    const int* src = ei;
    const int* dst = ei + E;

    char* ws = (char*)d_ws;
    size_t off = 0;
    auto alloc = [&](size_t bytes) -> void* {
        void* p = ws + off;
        off = (off + bytes + 255) & ~(size_t)255;
        return p;
    };
    unsigned int* deg    = (unsigned int*)alloc((size_t)N * sizeof(unsigned int));
    float*        dinv   = (float*)alloc((size_t)N * sizeof(float));
    float*        buf0   = (float*)alloc((size_t)N * HID * sizeof(float));
    float*        buf1   = (float*)alloc((size_t)N * HID * sizeof(float));
    float*        pooled = (float*)alloc((size_t)G * HID * sizeof(float));
    unsigned int* cnt    = (unsigned int*)alloc((size_t)G * sizeof(unsigned int));
    (void)ws_size; (void)n_in;

    const int T   = 256;
    const int gN  = (N + T - 1) / T;
    const int gE  = (E + T - 1) / T;
    const int gNH = (N * HID + T - 1) / T;
    const dim3 sb(64, 4);
    const int gS  = (E + 3) / 4;
    const int mt  = (N + 15) / 16;

    // normalization
    k_deg_init <<<gN, T, 0, stream>>>(deg, N);
    k_deg_count<<<gE, T, 0, stream>>>(deg, dst, E);
    k_dinv     <<<gN, T, 0, stream>>>(deg, dinv, N);

    // layer 1: h = x@W1 (buf0), agg = self-loop init (buf1), scatter, bias+relu -> buf0
    k_l1       <<<gNH, T, 0, stream>>>(x, W1, dinv, buf0, buf1, N);
    k_scatter  <<<gS, sb, 0, stream>>>(buf0, buf1, src, dst, dinv, E);
    k_bias_relu<<<gNH, T, 0, stream>>>(buf1, b1, buf0, N);

    // layer 2: h2pre = h1@W2 via WMMA (buf0 -> buf1), agg init -> buf0, scatter into buf0
    k_gemm64   <<<mt, 128, 0, stream>>>(buf0, W2, buf1, N);
    k_selfloop <<<gNH, T, 0, stream>>>(buf1, dinv, buf0, N);
    k_scatter  <<<gS, sb, 0, stream>>>(buf1, buf0, src, dst, dinv, E);

    // bias+relu fused with mean-pool accumulation, then head GEMV
    k_zero_pool<<<(G * HID + T - 1) / T, T, 0, stream>>>(pooled, cnt, G);
    k_relu_pool<<<gNH, T, 0, stream>>>(buf0, b2, batch, pooled, cnt, N);
    k_head     <<<(G + 127) / 128, 128, 0, stream>>>(pooled, cnt, Wfc, bfc, out, G);
}